// L1_ChamferEval_19164144075465
// MI455X (gfx1250) — compile-verified
//
#include <hip/hip_runtime.h>
#include <hip/hip_bf16.h>

typedef __attribute__((ext_vector_type(2))) float v2f;
typedef __attribute__((ext_vector_type(8))) float v8f;

#define B_  16
#define N_  4096
#define M_  4096
#define WAVES_PER_BLOCK 8
#define THREADS 256                                   // 8 wave32s
#define ROWBLOCKS_PER_BATCH (N_ / 16)                 // 256
#define BLOCKS_PER_BATCH (ROWBLOCKS_PER_BATCH / WAVES_PER_BLOCK)  // 32
#define TILES_PER_BATCH (M_ / 16)                     // 256
#define UNROLL 8                                      // 8 tiles = 128 columns / iter
#define FLT_BIG 3.402823466e+38f
#define INF_BITS 0x7F800000u

// ---------------------------------------------------------------------------
// Phase 0: pre-swizzle the B operand into WMMA wave32 register layout and
// precompute |y|^2; also init the global column-min buffer to +inf.
//   B 4x16 f32 layout: VGPR0 lanes0-15:K=0, lanes16-31:K=2
//                      VGPR1 lanes0-15:K=1, lanes16-31:K=3
// ypacked[b][tile][lane] (float2):
//   lane<16 : {-2y.x, -2y.y}   lane>=16 : {-2y.z, 1.0}
// ---------------------------------------------------------------------------
__global__ void chamfer_prep(const float* __restrict__ ys,
                             v2f* __restrict__ ypacked,
                             float* __restrict__ ynArr,
                             unsigned int* __restrict__ dist2bits) {
    int i = blockIdx.x * blockDim.x + threadIdx.x;    // one y point per thread
    if (i >= B_ * M_) return;
    int b = i / M_;
    int m = i % M_;
    const float* yp = ys + (size_t)i * 3;
    float y0 = yp[0], y1 = yp[1], y2 = yp[2];
    ynArr[i] = y0 * y0 + y1 * y1 + y2 * y2;
    dist2bits[i] = INF_BITS;
    int t = m >> 4, l = m & 15;
    size_t base = ((size_t)b * TILES_PER_BATCH + t) * 32;
    v2f lo; lo.x = -2.0f * y0; lo.y = -2.0f * y1;
    v2f hi; hi.x = -2.0f * y2; hi.y = 1.0f;
    ypacked[base + l]      = lo;
    ypacked[base + 16 + l] = hi;
}

// ---------------------------------------------------------------------------
// Phase 1: WMMA tile kernel. One wave owns 16 rows (x-points); per iteration
// it issues 8x{b64+b32} loads then 8x v_wmma_f32_16x16x4_f32 with C = inline
// 0, producing 128 columns of  A(16x4)*B(4x16) = |x|^2 - 2<x,y>.
// |y|^2 is added afterwards (quiet arithmetic results -> compiler can drop
// fmin canonicalization). Clamp-to-0 is applied only where it matters:
// once per tile on the column min (before the uint-bit atomic) and once at
// the end on the row-min accumulators (min and max(.,0) commute).
// ---------------------------------------------------------------------------
__global__ __launch_bounds__(THREADS)
void chamfer_tiles(const float* __restrict__ xs,
                   const v2f* __restrict__ ypacked,
                   const float* __restrict__ ynArr,
                   float* __restrict__ dist1,
                   unsigned int* __restrict__ dist2bits) {
    __shared__ unsigned int colmin[M_];               // 16 KB of 320 KB WGP LDS

    const int tid   = threadIdx.x;
    const int lane  = tid & 31;
    const int wave  = tid >> 5;
    const int batch = blockIdx.x / BLOCKS_PER_BATCH;
    const int rowBlock = (blockIdx.x % BLOCKS_PER_BATCH) * WAVES_PER_BLOCK + wave;
    const int rowBase  = rowBlock * 16;
    const int lr = lane & 15;

    for (int i = tid; i < M_; i += THREADS) colmin[i] = INF_BITS;
    __syncthreads();

    // --- A operand: this wave's 16 x-points, wave32 16x4 layout ---
    const float* xp = xs + ((size_t)batch * N_ + rowBase + lr) * 3;
    float x0 = xp[0], x1 = xp[1], x2 = xp[2];
    float xn = x0 * x0 + x1 * x1 + x2 * x2;
    v2f a;
    a.x = (lane < 16) ? x0 : x2;                      // K=0 / K=2
    a.y = (lane < 16) ? x1 : xn;                      // K=1 / K=3 (|x|^2)

    v8f rmin;
#pragma unroll
    for (int j = 0; j < 8; ++j) rmin[j] = FLT_BIG;

    const v2f*   ywave = ypacked + (size_t)batch * TILES_PER_BATCH * 32;
    const float* ynb   = ynArr + (size_t)batch * M_;

    for (int colBase = 0; colBase < M_; colBase += 16 * UNROLL) {
        const int tile0 = colBase >> 4;
        v2f   bt[UNROLL];
        float ynv[UNROLL];
#pragma unroll
        for (int t = 0; t < UNROLL; ++t) {            // 16 loads in flight
            bt[t]  = ywave[(size_t)(tile0 + t) * 32 + lane];
            ynv[t] = ynb[colBase + t * 16 + lr];
        }
#pragma unroll
        for (int t = 0; t < UNROLL; ++t) {
            v8f z = {};                               // C = inline 0 (no movs)
            // 8 args: (neg_a, A, neg_b, B, c_mod, C, reuse_a, reuse_b)
            v8f d = __builtin_amdgcn_wmma_f32_16x16x4_f32(
                false, a, false, bt[t], (short)0, z, false, false);

            float cm = FLT_BIG;
#pragma unroll
            for (int j = 0; j < 8; ++j) {
                float s  = d[j] + ynv[t];             // + |y|^2 (quiet result)
                rmin[j]  = fminf(rmin[j], s);
                cm       = fminf(cm, s);
            }
            cm = fmaxf(cm, 0.0f);                     // clamp before uint-bit min
            // lanes 0-15 carry rows 0-7 min, lanes 16-31 rows 8-15 min;
            // both halves fold into the same column slot.
            atomicMin(&colmin[colBase + t * 16 + lr], __float_as_uint(cm));
        }
    }

    // --- clamp + row-min cross-lane reduction within each 16-lane half ---
#pragma unroll
    for (int j = 0; j < 8; ++j) rmin[j] = fmaxf(rmin[j], 0.0f);
#pragma unroll
    for (int m = 8; m >= 1; m >>= 1) {
#pragma unroll
        for (int j = 0; j < 8; ++j)
            rmin[j] = fminf(rmin[j], __shfl_xor(rmin[j], m, 32));
    }
    if (lane == 0) {
#pragma unroll
        for (int j = 0; j < 8; ++j)
            dist1[(size_t)batch * N_ + rowBase + j] = rmin[j];
    } else if (lane == 16) {
#pragma unroll
        for (int j = 0; j < 8; ++j)
            dist1[(size_t)batch * N_ + rowBase + 8 + j] = rmin[j];
    }

    // --- flush block-local column mins (1 global atomic per col per block) ---
    __syncthreads();
    for (int i = tid; i < M_; i += THREADS)
        atomicMin(&dist2bits[(size_t)batch * M_ + i], colmin[i]);
}

// ---------------------------------------------------------------------------
// Phase 2: deterministic single-block final reduction (fixed order, no float
// atomics -> bit-identical across graph replays).
// ---------------------------------------------------------------------------
__global__ __launch_bounds__(1024)
void chamfer_final(const float* __restrict__ dist1, const float* __restrict__ dist2,
                   float* __restrict__ out) {
    __shared__ float s1[1024];
    __shared__ float s2[1024];
    float acc1 = 0.0f, acc2 = 0.0f;
    for (int i = threadIdx.x; i < B_ * N_; i += 1024) acc1 += sqrtf(dist1[i]);
    for (int i = threadIdx.x; i < B_ * M_; i += 1024) acc2 += sqrtf(dist2[i]);
    s1[threadIdx.x] = acc1;
    s2[threadIdx.x] = acc2;
    __syncthreads();
    for (int s = 512; s > 0; s >>= 1) {
        if (threadIdx.x < (unsigned)s) {
            s1[threadIdx.x] += s1[threadIdx.x + s];
            s2[threadIdx.x] += s2[threadIdx.x + s];
        }
        __syncthreads();
    }
    if (threadIdx.x == 0) {
        float mean1 = s1[0] / (float)(B_ * N_);
        float mean2 = s2[0] / (float)(B_ * M_);
        out[0] = (mean1 + mean2) * 0.5f * 1000.0f;
    }
}

// ---------------------------------------------------------------------------
extern "C" void kernel_launch(void* const* d_in, const int* in_sizes, int n_in,
                              void* d_out, int out_size, void* d_ws, size_t ws_size,
                              hipStream_t stream) {
    const float* xs = (const float*)d_in[0];          // [B, N, 3] f32
    const float* ys = (const float*)d_in[1];          // [B, M, 3] f32
    float* out = (float*)d_out;                       // scalar f32

    char* ws = (char*)d_ws;
    float*        dist1     = (float*)ws;                                   ws += (size_t)B_ * N_ * sizeof(float);
    unsigned int* dist2bits = (unsigned int*)ws;                            ws += (size_t)B_ * M_ * sizeof(unsigned int);
    float*        ynArr     = (float*)ws;                                   ws += (size_t)B_ * M_ * sizeof(float);
    v2f*          ypacked   = (v2f*)ws;   // B * (M/16) * 32 * float2 = 1 MB

    chamfer_prep<<<(B_ * M_ + 255) / 256, 256, 0, stream>>>(
        ys, ypacked, ynArr, dist2bits);

    chamfer_tiles<<<B_ * BLOCKS_PER_BATCH, THREADS, 0, stream>>>(
        xs, ypacked, ynArr, dist1, dist2bits);

    chamfer_final<<<1, 1024, 0, stream>>>(dist1, (const float*)dist2bits, out);
}